// HGConvLayer_67568425500743
// MI455X (gfx1250) — compile-verified
//
#include <hip/hip_runtime.h>
#include <math.h>

// ---------------- problem constants ----------------
#define KH   8          // heads
#define DH   32         // dim per head
#define KD   256        // KH*DH
#define NA   50000
#define NP   100000
#define DIN  128
#define SLOPE 0.2f

// ---------------- WMMA types ----------------
typedef float v2f __attribute__((ext_vector_type(2)));
typedef float v8f __attribute__((ext_vector_type(8)));

// =====================================================================
// Dense GEMM: C[M x 256] = act(A[M x Kdim]) @ B[Kdim x 256]
// f32 WMMA 16x16x4.  block = 256 threads = 8 waves, each wave owns a
// 16x32 strip of C (two 16x16 accumulators sharing the A fragment).
// grid.x = M/16 (M is always a multiple of 16 here).
// A-frag layout (16x4 f32): lanes 0-15 hold K=kb,kb+1; lanes 16-31 K=kb+2,kb+3.
// B-frag (4x16): VGPR0 = rows kb (lanes0-15) / kb+2 (lanes16-31), VGPR1 = kb+1/kb+3.
// C (16x16 f32): VGPR i -> row i (lanes0-15), row i+8 (lanes16-31).
// =====================================================================
__global__ void __launch_bounds__(256)
hg_gemm_wmma_f32(const float* __restrict__ A, const float* __restrict__ B,
                 float* __restrict__ C, int Kdim, int reluA) {
    const int wave = threadIdx.x >> 5;
    const int lane = threadIdx.x & 31;
    const int half = lane >> 4;
    const int lrow = lane & 15;
    const int m0   = blockIdx.x * 16;
    const int n0   = wave * 32;

    const float* __restrict__ arow = A + (size_t)(m0 + lrow) * (size_t)Kdim;

    v8f c0 = {};
    v8f c1 = {};

    for (int kb = 0; kb < Kdim; kb += 4) {
        const int ka = kb + 2 * half;
        v2f a;
        a.x = arow[ka];
        a.y = arow[ka + 1];
        if (reluA) { a.x = fmaxf(a.x, 0.0f); a.y = fmaxf(a.y, 0.0f); }

        const float* __restrict__ brow = B + (size_t)ka * KD;
        v2f b0, b1;
        b0.x = brow[n0 + lrow];
        b0.y = brow[KD + n0 + lrow];
        b1.x = brow[n0 + 16 + lrow];
        b1.y = brow[KD + n0 + 16 + lrow];

        c0 = __builtin_amdgcn_wmma_f32_16x16x4_f32(false, a, false, b0,
                                                   (short)0, c0, false, false);
        c1 = __builtin_amdgcn_wmma_f32_16x16x4_f32(false, a, false, b1,
                                                   (short)0, c1, false, false);
    }

    float* __restrict__ crow = C + (size_t)(m0 + 8 * half) * KD + n0 + lrow;
#pragma unroll
    for (int i = 0; i < 8; ++i) {
        crow[(size_t)i * KD]      = c0[i];
        crow[(size_t)i * KD + 16] = c1[i];
    }
}

// =====================================================================
// Utility / sparse kernels
// =====================================================================
__global__ void hg_fill_f32(float* __restrict__ p, float v, long long n) {
    long long i = (long long)blockIdx.x * blockDim.x + threadIdx.x;
    if (i < n) p[i] = v;
}

__device__ __forceinline__ float hg_leaky(float x) {
    return x >= 0.0f ? x : SLOPE * x;
}

__device__ __forceinline__ void hg_atomic_max_f32(float* addr, float val) {
    // sign-aware trick; buffer initialized to -inf (0xFF800000)
    if (val >= 0.0f) atomicMax((int*)addr, __float_as_int(val));
    else             atomicMin((unsigned int*)addr, __float_as_uint(val));
}

// per-(node, head) attention projection: out[n,k] = sum_d f[n,k,d]*attn[k, off+d]
__global__ void hg_node_score(const float* __restrict__ f,
                              const float* __restrict__ attn, int off,
                              float* __restrict__ out, int N) {
    int i = blockIdx.x * blockDim.x + threadIdx.x;
    if (i >= N * KH) return;
    const int n = i >> 3, k = i & 7;
    const float* fr = f + (size_t)n * KD + k * DH;
    const float* ar = attn + k * (2 * DH) + off;
    float s = 0.0f;
#pragma unroll
    for (int d = 0; d < DH; ++d) s += fr[d] * ar[d];
    out[i] = s;
}

// pass 1: e = leaky(el[src]+er[dst]); segment max over dst
__global__ void hg_edge_max(const float* __restrict__ el, const float* __restrict__ er,
                            const int* __restrict__ src, const int* __restrict__ dst,
                            float* __restrict__ e, float* __restrict__ m, int E) {
    long long i = (long long)blockIdx.x * blockDim.x + threadIdx.x;
    if (i >= (long long)E * KH) return;
    const int eid = (int)(i >> 3), k = (int)(i & 7);
    const int sn = src[eid], dn = dst[eid];
    float x = hg_leaky(el[(size_t)sn * KH + k] + er[(size_t)dn * KH + k]);
    e[i] = x;
    hg_atomic_max_f32(&m[(size_t)dn * KH + k], x);
}

// pass 2: s[dst,k] += exp(e - m[dst,k])
__global__ void hg_edge_sum(const float* __restrict__ e, const float* __restrict__ m,
                            const int* __restrict__ dst, float* __restrict__ s, int E) {
    long long i = (long long)blockIdx.x * blockDim.x + threadIdx.x;
    if (i >= (long long)E * KH) return;
    const int eid = (int)(i >> 3), k = (int)(i & 7);
    const size_t hk = (size_t)dst[eid] * KH + k;
    atomicAdd(&s[hk], __expf(e[i] - m[hk]));
}

// pass 3: out[dst, :] += fs[src, :] * exp(e-m)/s   (one block per edge)
__global__ void __launch_bounds__(256)
hg_edge_scatter(const float* __restrict__ fs, const float* __restrict__ e,
                const float* __restrict__ m, const float* __restrict__ s,
                const int* __restrict__ src, const int* __restrict__ dst,
                float* __restrict__ out) {
    const int eid = blockIdx.x;
    const int f   = threadIdx.x;          // 0..255
    const int k   = f >> 5;
    const int sn = src[eid], dn = dst[eid];
    const size_t hk = (size_t)dn * KH + k;
    const float w = __expf(e[(size_t)eid * KH + k] - m[hk]) / s[hk];
    atomicAdd(&out[(size_t)dn * KD + f], fs[(size_t)sn * KD + f] * w);
}

// author epilogue: out_a = sig(rw)*rf_wb + (1-sig(rw))*(res_a + b)
__global__ void hg_author_out(const float* __restrict__ rfwb,
                              const float* __restrict__ resa,
                              const float* __restrict__ bres,
                              const float* __restrict__ resw,
                              float* __restrict__ outa, long long n) {
    long long i = (long long)blockIdx.x * blockDim.x + threadIdx.x;
    if (i >= n) return;
    const int f = (int)(i & (KD - 1));
    const float aa = 1.0f / (1.0f + __expf(-resw[0]));
    outa[i] = aa * rfwb[i] + (1.0f - aa) * (resa[i] + bres[f]);
}

// paper epilogue: macro attention over {writes, cites} + residual
__global__ void hg_paper_out(const float* __restrict__ nodep,
                             const float* __restrict__ rfw,
                             const float* __restrict__ rfc,
                             const float* __restrict__ am,
                             const float* __restrict__ resp,
                             const float* __restrict__ bres,
                             const float* __restrict__ resw,
                             float* __restrict__ outp, int N) {
    int i = blockIdx.x * blockDim.x + threadIdx.x;
    if (i >= N * KH) return;
    const int n = i >> 3, k = i & 7;
    const float* a0 = am + k * (2 * DH);
    const float* pn = nodep + (size_t)n * KD + k * DH;
    const float* pw = rfw   + (size_t)n * KD + k * DH;
    const float* pc = rfc   + (size_t)n * KD + k * DH;

    float sw = 0.0f, sc = 0.0f;
#pragma unroll
    for (int d = 0; d < DH; ++d) {
        const float cn = a0[d] * pn[d];
        sw += cn + a0[DH + d] * pw[d];
        sc += cn + a0[DH + d] * pc[d];
    }
    sw = hg_leaky(sw);
    sc = hg_leaky(sc);
    const float mx = fmaxf(sw, sc);
    const float ew = __expf(sw - mx), ec = __expf(sc - mx);
    const float inv = 1.0f / (ew + ec);
    const float ww = ew * inv, wc = ec * inv;

    const float ap = 1.0f / (1.0f + __expf(-resw[0]));
    const float* rp = resp + (size_t)n * KD + k * DH;
    const float* bp = bres + k * DH;
    float* o = outp + (size_t)n * KD + k * DH;
#pragma unroll
    for (int d = 0; d < DH; ++d) {
        const float agg = ww * pw[d] + wc * pc[d];
        o[d] = ap * agg + (1.0f - ap) * (rp[d] + bp[d]);
    }
}

// =====================================================================
// host orchestration
// =====================================================================
extern "C" void kernel_launch(void* const* d_in, const int* in_sizes, int n_in,
                              void* d_out, int out_size, void* d_ws, size_t ws_size,
                              hipStream_t stream) {
    const float* fea   = (const float*)d_in[0];   // NA x 128
    const float* fep   = (const float*)d_in[1];   // NP x 128
    const float* Wma   = (const float*)d_in[2];   // 128 x 256
    const float* Wmp   = (const float*)d_in[3];
    const float* amA   = (const float*)d_in[4];   // 8 x 64
    const float* amP   = (const float*)d_in[5];
    /* d_in[6] = W_macro_node_author : unused by reference */
    const float* Wmnp  = (const float*)d_in[7];
    const float* Wrw   = (const float*)d_in[8];   // 256 x 256
    const float* Wrwb  = (const float*)d_in[9];
    const float* Wrc   = (const float*)d_in[10];
    const float* amM   = (const float*)d_in[11];  // 8 x 64
    const float* WresA = (const float*)d_in[12];
    const float* bresA = (const float*)d_in[13];
    const float* WresP = (const float*)d_in[14];
    const float* bresP = (const float*)d_in[15];
    const float* rwA   = (const float*)d_in[16];
    const float* rwP   = (const float*)d_in[17];
    const int* w_src  = (const int*)d_in[18];
    const int* w_dst  = (const int*)d_in[19];
    const int* wb_src = (const int*)d_in[20];
    const int* wb_dst = (const int*)d_in[21];
    const int* c_src  = (const int*)d_in[22];
    const int* c_dst  = (const int*)d_in[23];
    const int Ew  = in_sizes[18];
    const int Ewb = in_sizes[20];
    const int Ec  = in_sizes[22];

    // ------- workspace layout (floats) -------
    const size_t NAF = (size_t)NA * KD;   // 12.8M
    const size_t NPF = (size_t)NP * KD;   // 25.6M
    float* w  = (float*)d_ws;
    float* S0 = w;           // fa        -> later rf_wb
    float* S1 = S0 + NAF;    // fp        -> later rf_writes
    float* S2 = S1 + NPF;    // r_writes  -> later rf_cites
    float* S3 = S2 + NPF;    // r_wb      -> later res_a
    float* S4 = S3 + NAF;    // r_cites   -> later node_p
    float* S5 = S4 + NPF;    // res_p
    float* EL = S5 + NPF;                     // max NP*8
    float* ER = EL + (size_t)NP * KH;
    float* Mx = ER + (size_t)NP * KH;
    float* Sx = Mx + (size_t)NP * KH;
    float* Ee = Sx + (size_t)NP * KH;         // max Ec*8 = 8M

    float* out_a = (float*)d_out;             // NA x 256
    float* out_p = out_a + NAF;               // NP x 256

    auto fill = [&](float* p, float v, long long n) {
        hg_fill_f32<<<(unsigned)((n + 255) / 256), 256, 0, stream>>>(p, v, n);
    };
    auto gemm = [&](const float* A, const float* B, float* C, int M, int Kdim, int relu) {
        hg_gemm_wmma_f32<<<M / 16, 256, 0, stream>>>(A, B, C, Kdim, relu);
    };
    auto micro = [&](const float* fs, const float* fd, const float* attn,
                     const int* src, const int* dst, int E, int Nsrc, int Ndst,
                     float* out) {
        hg_node_score<<<(Nsrc * KH + 255) / 256, 256, 0, stream>>>(fs, attn, 0,  EL, Nsrc);
        hg_node_score<<<(Ndst * KH + 255) / 256, 256, 0, stream>>>(fd, attn, DH, ER, Ndst);
        fill(Mx, -INFINITY, (long long)Ndst * KH);
        fill(Sx, 0.0f,      (long long)Ndst * KH);
        fill(out, 0.0f,     (long long)Ndst * KD);
        const unsigned gE = (unsigned)(((long long)E * KH + 255) / 256);
        hg_edge_max<<<gE, 256, 0, stream>>>(EL, ER, src, dst, Ee, Mx, E);
        hg_edge_sum<<<gE, 256, 0, stream>>>(Ee, Mx, dst, Sx, E);
        hg_edge_scatter<<<E, 256, 0, stream>>>(fs, Ee, Mx, Sx, src, dst, out);
    };

    // 1-2: micro projections
    gemm(fea, Wma, S0, NA, DIN, 0);            // fa
    gemm(fep, Wmp, S1, NP, DIN, 0);            // fp
    // 3-5: three GAT relations
    micro(S0, S1, amA, w_src,  w_dst,  Ew,  NA, NP, S2);   // r_writes (papers)
    micro(S1, S0, amP, wb_src, wb_dst, Ewb, NP, NA, S3);   // r_wb (authors)
    micro(S1, S1, amP, c_src,  c_dst,  Ec,  NP, NP, S4);   // r_cites (papers)
    // 6-8: relation transforms with fused ReLU on the input
    gemm(S2, Wrw,  S1, NP, KD, 1);             // rf_writes (reuse fp slot)
    gemm(S3, Wrwb, S0, NA, KD, 1);             // rf_wb     (reuse fa slot)
    gemm(S4, Wrc,  S2, NP, KD, 1);             // rf_cites  (reuse r_writes slot)
    // 9-11: macro node proj + residual projections
    gemm(fep, Wmnp,  S4, NP, DIN, 0);          // node_p    (reuse r_cites slot)
    gemm(fea, WresA, S3, NA, DIN, 0);          // res_a     (reuse r_wb slot)
    gemm(fep, WresP, S5, NP, DIN, 0);          // res_p
    // 12: author output (gated residual)
    hg_author_out<<<(unsigned)((NAF + 255) / 256), 256, 0, stream>>>(
        S0, S3, bresA, rwA, out_a, (long long)NAF);
    // 13: paper output (macro attention softmax + gated residual)
    hg_paper_out<<<(NP * KH + 255) / 256, 256, 0, stream>>>(
        S4, S1, S2, amM, S5, bresP, rwP, out_p, NP);
}